// MultiHeadGATLayer_22239340659366
// MI455X (gfx1250) — compile-verified
//
#include <hip/hip_runtime.h>
#include <math.h>

// ---------------------------------------------------------------------------
// MultiHeadGAT on gfx1250 (MI455X).
// N = 100000 nodes, E = 600000 edges, IN = 128, H = 8 heads, D = 16.
// z = h@W via V_WMMA_F32_16X16X4_F32 (fp32 WMMA: full reference precision,
// GEMM is <5% of runtime so no reason to quantize). Edge phase is gather /
// scatter bandwidth: 3 edge passes with f32 atomics (max, add, add).
// ---------------------------------------------------------------------------

typedef float v2f __attribute__((ext_vector_type(2)));
typedef float v8f __attribute__((ext_vector_type(8)));

#define NEG_SLOPE 0.2f

// float atomic max via int-max / uint-min trick (valid with -inf init).
__device__ __forceinline__ void atomicMaxF(float* addr, float val) {
    if (val >= 0.0f) atomicMax((int*)addr, __float_as_int(val));
    else             atomicMin((unsigned int*)addr, __float_as_uint(val));
}

// ---------------------------------------------------------------------------
// 0) init: out (agg accumulator) = 0, m = -inf, s = 0
// ---------------------------------------------------------------------------
__global__ __launch_bounds__(256) void k_init(float* __restrict__ out,
                                              float* __restrict__ m,
                                              float* __restrict__ s,
                                              int n_out, int n_nh) {
    int i = blockIdx.x * 256 + threadIdx.x;
    if (i < n_out) out[i] = 0.0f;
    if (i < n_nh) { m[i] = -INFINITY; s[i] = 0.0f; }
}

// ---------------------------------------------------------------------------
// 1) z = h @ W.  Block = 256 thr = 8 waves. Block b: rows [16b,16b+16).
//    Wave w: cols [16w,16w+16) == head w. 32 x WMMA_F32_16X16X4_F32 (K=128).
//    A frag (16x4 f32): lanes 0-15 hold M=lane, v0=K0,v1=K1; lanes 16-31 K2,K3.
//    B frag (4x16 f32): lanes 0-15 hold N=lane, v0=K0,v1=K1; lanes 16-31 K2,K3.
//    C/D: vgpr v, lane -> (M = v + 8*(lane>=16), N = lane&15).
// ---------------------------------------------------------------------------
__global__ __launch_bounds__(256) void k_gemm_wmma(const float* __restrict__ h,
                                                   const float* __restrict__ W,
                                                   float* __restrict__ z,
                                                   int N) {
    const int lane    = threadIdx.x & 31;
    const int wave    = threadIdx.x >> 5;        // 0..7 (== head)
    const int rowBase = blockIdx.x * 16;
    if (rowBase >= N) return;                    // uniform per wave

    const int laneLo  = lane & 15;
    const int hi      = lane >> 4;               // 0: K0/K1  1: K2/K3
    const int colBase = wave * 16;

    int arow = rowBase + laneLo;                 // clamp so EXEC stays full
    if (arow >= N) arow = N - 1;
    const int bcol = colBase + laneLo;

    const float* __restrict__ aptr = h + (size_t)arow * 128 + hi * 2;
    const float* __restrict__ bptr = W + (size_t)(hi * 2) * 128 + bcol;

    v8f acc = {};
#pragma unroll 8
    for (int k4 = 0; k4 < 32; ++k4) {
        v2f a, b;
        float2 av = *(const float2*)(aptr + k4 * 4);   // 8B aligned: kk even
        a.x = av.x;
        a.y = av.y;
        b.x = bptr[k4 * 4 * 128];                      // W[kk    ][col]
        b.y = bptr[k4 * 4 * 128 + 128];                // W[kk + 1][col]
        acc = __builtin_amdgcn_wmma_f32_16x16x4_f32(
            /*neg_a=*/false, a, /*neg_b=*/false, b,
            /*c_mod=*/(short)0, acc, /*reuse_a=*/false, /*reuse_b=*/false);
    }

    // Store: uniform fast path (full 16-row tile) avoids 8x exec-mask
    // branches; N % 16 == 0 here so this is the only path actually taken.
    if (rowBase + 16 <= N) {
        float* __restrict__ zp =
            z + (size_t)(rowBase + hi * 8) * 128 + colBase + laneLo;
#pragma unroll
        for (int v = 0; v < 8; ++v) zp[(size_t)v * 128] = acc[v];
    } else {
#pragma unroll
        for (int v = 0; v < 8; ++v) {
            int row = rowBase + v + hi * 8;
            if (row < N) z[(size_t)row * 128 + colBase + laneLo] = acc[v];
        }
    }
}

// ---------------------------------------------------------------------------
// 2) el[n,h] = <z[n,h,:], attn_l[h,:]>, er likewise. One thread per (n,head).
// ---------------------------------------------------------------------------
__global__ __launch_bounds__(256) void k_logits(const float* __restrict__ z,
                                                const float* __restrict__ attn_l,
                                                const float* __restrict__ attn_r,
                                                float* __restrict__ el,
                                                float* __restrict__ er,
                                                int NH) {
    int t = blockIdx.x * 256 + threadIdx.x;
    if (t >= NH) return;
    int head = t & 7;
    int n    = t >> 3;
    const float* __restrict__ zp = z + (size_t)n * 128 + head * 16;
    const float* __restrict__ al = attn_l + head * 16;
    const float* __restrict__ ar = attn_r + head * 16;
    float sl = 0.0f, sr = 0.0f;
#pragma unroll
    for (int j = 0; j < 16; ++j) {
        float v = zp[j];
        sl = fmaf(v, al[j], sl);
        sr = fmaf(v, ar[j], sr);
    }
    el[t] = sl;
    er[t] = sr;
}

// ---------------------------------------------------------------------------
// 3) edge pass 1: e = leaky(el[dst]+er[src]); segment max into m[dst].
// ---------------------------------------------------------------------------
__global__ __launch_bounds__(256) void k_edge_max(const long long* __restrict__ src,
                                                  const long long* __restrict__ dst,
                                                  const float* __restrict__ el,
                                                  const float* __restrict__ er,
                                                  float* __restrict__ m, int E) {
    int e = blockIdx.x * 256 + threadIdx.x;
    if (e >= E) return;
    long long s = src[e], d = dst[e];
    const float* __restrict__ elp = el + d * 8;
    const float* __restrict__ erp = er + s * 8;
    float* __restrict__ mp = m + d * 8;
#pragma unroll
    for (int h0 = 0; h0 < 8; ++h0) {
        float v = elp[h0] + erp[h0];
        v = v > 0.0f ? v : NEG_SLOPE * v;
        atomicMaxF(&mp[h0], v);
    }
}

// ---------------------------------------------------------------------------
// 4) edge pass 2: ex = exp(e - m[dst]); segment sum into s[dst].
// ---------------------------------------------------------------------------
__global__ __launch_bounds__(256) void k_edge_sum(const long long* __restrict__ src,
                                                  const long long* __restrict__ dst,
                                                  const float* __restrict__ el,
                                                  const float* __restrict__ er,
                                                  const float* __restrict__ m,
                                                  float* __restrict__ ssum, int E) {
    int e = blockIdx.x * 256 + threadIdx.x;
    if (e >= E) return;
    long long s = src[e], d = dst[e];
    const float* __restrict__ elp = el + d * 8;
    const float* __restrict__ erp = er + s * 8;
    const float* __restrict__ mp  = m + d * 8;
    float* __restrict__ sp = ssum + d * 8;
#pragma unroll
    for (int h0 = 0; h0 < 8; ++h0) {
        float v = elp[h0] + erp[h0];
        v = v > 0.0f ? v : NEG_SLOPE * v;
        atomicAdd(&sp[h0], expf(v - mp[h0]));
    }
}

// ---------------------------------------------------------------------------
// 5) edge pass 3: one wave per edge. Lane l owns float4 at offset 4l of the
//    128-wide feature (head = l/4). agg[dst] += alpha * z[src] via f32 atomics.
// ---------------------------------------------------------------------------
__global__ __launch_bounds__(256) void k_edge_agg(const long long* __restrict__ src,
                                                  const long long* __restrict__ dst,
                                                  const float* __restrict__ el,
                                                  const float* __restrict__ er,
                                                  const float* __restrict__ m,
                                                  const float* __restrict__ ssum,
                                                  const float* __restrict__ z,
                                                  float* __restrict__ out, int E) {
    int lane = threadIdx.x & 31;
    int e    = blockIdx.x * 8 + (threadIdx.x >> 5);
    if (e >= E) return;
    long long s = src[e], d = dst[e];
    int head = lane >> 2;
    float v = el[d * 8 + head] + er[s * 8 + head];
    v = v > 0.0f ? v : NEG_SLOPE * v;
    float alpha = expf(v - m[d * 8 + head]) / ssum[d * 8 + head];

    float4 zv = *(const float4*)(z + (size_t)s * 128 + lane * 4);
    float* __restrict__ op = out + (size_t)d * 128 + lane * 4;
    atomicAdd(op + 0, alpha * zv.x);
    atomicAdd(op + 1, alpha * zv.y);
    atomicAdd(op + 2, alpha * zv.z);
    atomicAdd(op + 3, alpha * zv.w);
}

// ---------------------------------------------------------------------------
// 6) finalize: out = o + elu(agg)   (agg accumulated in-place in d_out)
// ---------------------------------------------------------------------------
__global__ __launch_bounds__(256) void k_final(const float* __restrict__ o,
                                               float* __restrict__ out, int total) {
    int i = blockIdx.x * 256 + threadIdx.x;
    if (i >= total) return;
    float x = out[i];
    float e = x > 0.0f ? x : (expf(x) - 1.0f);
    out[i] = o[i] + e;
}

// ---------------------------------------------------------------------------
extern "C" void kernel_launch(void* const* d_in, const int* in_sizes, int n_in,
                              void* d_out, int out_size, void* d_ws, size_t ws_size,
                              hipStream_t stream) {
    const float*     h      = (const float*)d_in[0];
    const float*     o      = (const float*)d_in[1];
    const float*     W      = (const float*)d_in[2];
    const float*     attn_l = (const float*)d_in[3];
    const float*     attn_r = (const float*)d_in[4];
    const long long* src    = (const long long*)d_in[5];
    const long long* dst    = (const long long*)d_in[6];

    const int N  = in_sizes[0] / 128;     // 100000
    const int E  = in_sizes[5];           // 600000
    const int NH = N * 8;
    const int NF = N * 128;

    float* out = (float*)d_out;

    // workspace layout: z | el | er | m | s   (N*160 floats = 64 MB)
    float* z  = (float*)d_ws;
    float* el = z  + (size_t)N * 128;
    float* er = el + (size_t)NH;
    float* m  = er + (size_t)NH;
    float* ss = m  + (size_t)NH;

    const int T = 256;

    k_init<<<(NF + T - 1) / T, T, 0, stream>>>(out, m, ss, NF, NH);
    k_gemm_wmma<<<(N + 15) / 16, T, 0, stream>>>(h, W, z, N);
    k_logits<<<(NH + T - 1) / T, T, 0, stream>>>(z, attn_l, attn_r, el, er, NH);
    k_edge_max<<<(E + T - 1) / T, T, 0, stream>>>(src, dst, el, er, m, E);
    k_edge_sum<<<(E + T - 1) / T, T, 0, stream>>>(src, dst, el, er, m, ss, E);
    k_edge_agg<<<(E + 7) / 8, T, 0, stream>>>(src, dst, el, er, m, ss, z, out, E);
    k_final<<<(NF + T - 1) / T, T, 0, stream>>>(o, out, NF);
}